// GAT_9208409883098
// MI455X (gfx1250) — compile-verified
//
#include <hip/hip_runtime.h>
#include <math.h>

// ---- problem constants (from reference) ----
#define IN_C   128
#define HID    32
#define HEADS  4
#define OUT_C  64
#define F1     (HEADS * HID)   // 128
#define NEG_SLOPE 0.2f

typedef float v2f __attribute__((ext_vector_type(2)));
typedef float v8f __attribute__((ext_vector_type(8)));

// ------------------------------------------------------------------
// float atomic max via CAS (order-independent result)
// ------------------------------------------------------------------
__device__ __forceinline__ void atomicMaxF(float* addr, float val) {
    unsigned int* ai = reinterpret_cast<unsigned int*>(addr);
    unsigned int old = __float_as_uint(*addr);
    while (__uint_as_float(old) < val) {
        unsigned int assumed = old;
        old = atomicCAS(ai, assumed, __float_as_uint(val));
        if (old == assumed) break;
    }
}

// ------------------------------------------------------------------
// fill kernel (re-init scratch each launch; graph-capture safe)
// ------------------------------------------------------------------
__global__ void gat_fill(float* __restrict__ p, float v, int n) {
    int i = blockIdx.x * blockDim.x + threadIdx.x;
    if (i < n) p[i] = v;
}

// ------------------------------------------------------------------
// C[M x Ncols] = A[M x K] @ B[K x Ncols], fp32 via V_WMMA_F32_16X16X4_F32
// block = 32 threads (one wave32), grid = (M/16, Ncols/(16*NT))
// Each wave computes a 16 x (16*NT) strip: one A fragment per k-step is
// shared (in registers) by NT back-to-back WMMAs into NT accumulators.
// A-frag (16x4): lane m = lane&15; VGPR0/1 hold K={0,1} (lanes 0-15)
//                or K={2,3} (lanes 16-31)               [ISA 7.12.2]
// B-frag (4x16): lane n = lane&15; same K split across lane halves
// C/D (16x16):   VGPR v holds row m = v + 8*(lane>=16), col = lane&15
// ------------------------------------------------------------------
template <int NT>
__global__ void gat_gemm_wmma_f32(const float* __restrict__ A,
                                  const float* __restrict__ B,
                                  float* __restrict__ C,
                                  int K, int Ncols) {
    const int lane = threadIdx.x;       // 0..31
    const int half = lane >> 4;         // 0 or 1
    const int l16  = lane & 15;
    const int row     = blockIdx.x * 16 + l16;
    const int colbase = blockIdx.y * (16 * NT);

    const float* __restrict__ arow = A + (long)row * K;
    const float* __restrict__ bcol = B + colbase + l16;

    v8f acc[NT];
#pragma unroll
    for (int t = 0; t < NT; ++t) acc[t] = (v8f){};

    for (int k = 0; k < K; k += 4) {
        const int ka = k + 2 * half;                 // even -> 8B aligned
        const v2f a = *(const v2f*)(arow + ka);      // global_load_b64
        const float* __restrict__ b0 = bcol + (long)ka * Ncols;
        const float* __restrict__ b1 = b0 + Ncols;
#pragma unroll
        for (int t = 0; t < NT; ++t) {
            v2f b;
            b.x = b0[t * 16];
            b.y = b1[t * 16];
            acc[t] = __builtin_amdgcn_wmma_f32_16x16x4_f32(
                         /*neg_a=*/false, a, /*neg_b=*/false, b,
                         /*c_mod=*/(short)0, acc[t],
                         /*reuse_a=*/false, /*reuse_b=*/false);
        }
    }

#pragma unroll
    for (int t = 0; t < NT; ++t) {
#pragma unroll
        for (int v = 0; v < 8; ++v) {
            const int m = blockIdx.x * 16 + v + 8 * half;
            C[(long)m * Ncols + colbase + t * 16 + l16] = acc[t][v];
        }
    }
}

// ------------------------------------------------------------------
// per-(node,head) attention dot products:
//   asrc[n,h] = sum_c h[n,h,c]*a_src[h,c] ; adst likewise
// ------------------------------------------------------------------
__global__ void gat_node_alpha(const float* __restrict__ hfeat,
                               const float* __restrict__ a_src,
                               const float* __restrict__ a_dst,
                               float* __restrict__ asrc,
                               float* __restrict__ adst,
                               int n, int H, int C) {
    int idx = blockIdx.x * blockDim.x + threadIdx.x;
    if (idx >= n * H) return;
    int node = idx / H;
    int hh   = idx - node * H;
    const float* row = hfeat + (long)node * (H * C) + hh * C;
    const float* ws_ = a_src + hh * C;
    const float* wd_ = a_dst + hh * C;
    float s = 0.f, d = 0.f;
    for (int c = 0; c < C; ++c) {
        float v = row[c];
        s += v * ws_[c];
        d += v * wd_[c];
    }
    asrc[idx] = s;
    adst[idx] = d;
}

// ------------------------------------------------------------------
// edge helpers: edges 0..E-1 come from edge_index (row0=src,row1=dst),
// edges E..E+N-1 are self loops
// ------------------------------------------------------------------
__device__ __forceinline__ void edge_sd(const int* __restrict__ ei, int E, int e,
                                        int& s, int& d) {
    if (e < E) { s = ei[e]; d = ei[E + e]; }
    else       { s = e - E; d = e - E; }
}

// pass 1: segment max of leaky_relu(asrc[s]+adst[d]) over dst
__global__ void gat_edge_max(const int* __restrict__ ei, int E, int n,
                             const float* __restrict__ asrc,
                             const float* __restrict__ adst,
                             float* __restrict__ amax, int H) {
    int e = blockIdx.x * blockDim.x + threadIdx.x;
    int EE = E + n;
    if (e >= EE) return;
    int s, d; edge_sd(ei, E, e, s, d);
    for (int h = 0; h < H; ++h) {
        float a = asrc[s * H + h] + adst[d * H + h];
        a = (a > 0.f) ? a : NEG_SLOPE * a;
        atomicMaxF(&amax[d * H + h], a);
    }
}

// pass 2: denom[d] += exp(a - amax[d])
__global__ void gat_edge_sum(const int* __restrict__ ei, int E, int n,
                             const float* __restrict__ asrc,
                             const float* __restrict__ adst,
                             const float* __restrict__ amax,
                             float* __restrict__ den, int H) {
    int e = blockIdx.x * blockDim.x + threadIdx.x;
    int EE = E + n;
    if (e >= EE) return;
    int s, d; edge_sd(ei, E, e, s, d);
    for (int h = 0; h < H; ++h) {
        float a = asrc[s * H + h] + adst[d * H + h];
        a = (a > 0.f) ? a : NEG_SLOPE * a;
        atomicAdd(&den[d * H + h], __expf(a - amax[d * H + h]));
    }
}

// pass 3: acc[d, :] += h[s, :] * softmax_weight     (one wave per edge,
// lane = feature channel -> coalesced 128B gathers of h[s])
__global__ void gat_edge_scatter(const int* __restrict__ ei, int E, int n,
                                 const float* __restrict__ asrc,
                                 const float* __restrict__ adst,
                                 const float* __restrict__ amax,
                                 const float* __restrict__ den,
                                 const float* __restrict__ hfeat,
                                 float* __restrict__ acc, int H, int C) {
    const int F = H * C;
    int warp = (blockIdx.x * blockDim.x + threadIdx.x) >> 5;
    int lane = threadIdx.x & 31;
    int EE = E + n;
    if (warp >= EE) return;
    int s, d; edge_sd(ei, E, warp, s, d);
    const float* hrow = hfeat + (long)s * F;
    float*       arow = acc   + (long)d * F;
    for (int c = lane; c < F; c += 32) {
        int h = c / C;
        float a = asrc[s * H + h] + adst[d * H + h];
        a = (a > 0.f) ? a : NEG_SLOPE * a;
        float w = __expf(a - amax[d * H + h]) / den[d * H + h];
        atomicAdd(&arow[c], hrow[c] * w);
    }
}

// elu(acc + b1) in place (layer-1 activation, becomes layer-2 input)
__global__ void gat_elu_bias(float* __restrict__ acc, const float* __restrict__ b,
                             int n, int F) {
    int i = blockIdx.x * blockDim.x + threadIdx.x;
    if (i >= n * F) return;
    float v = acc[i] + b[i % F];
    acc[i] = (v > 0.f) ? v : expm1f(v);
}

// out = acc + b2 (heads==1 so mean over heads is identity)
__global__ void gat_final_bias(const float* __restrict__ acc,
                               const float* __restrict__ b,
                               float* __restrict__ out, int n, int F) {
    int i = blockIdx.x * blockDim.x + threadIdx.x;
    if (i >= n * F) return;
    out[i] = acc[i] + b[i % F];
}

// ------------------------------------------------------------------
// launch
// ------------------------------------------------------------------
extern "C" void kernel_launch(void* const* d_in, const int* in_sizes, int n_in,
                              void* d_out, int out_size, void* d_ws, size_t ws_size,
                              hipStream_t stream) {
    const float* x        = (const float*)d_in[0];   // [N,128]
    const int*   ei       = (const int*)  d_in[1];   // [2,E]
    const float* W1       = (const float*)d_in[2];   // [128,128]
    const float* att_src1 = (const float*)d_in[3];   // [4,32]
    const float* att_dst1 = (const float*)d_in[4];   // [4,32]
    const float* b1       = (const float*)d_in[5];   // [128]
    const float* W2       = (const float*)d_in[6];   // [128,64]
    const float* att_src2 = (const float*)d_in[7];   // [1,64]
    const float* att_dst2 = (const float*)d_in[8];   // [1,64]
    const float* b2       = (const float*)d_in[9];   // [64]
    float* out = (float*)d_out;

    const int N  = in_sizes[0] / IN_C;    // 50000
    const int E  = in_sizes[1] / 2;       // 800000
    const int EE = E + N;                 // with self loops

    // ---- workspace layout (floats) ----
    float* ws = (float*)d_ws;
    float* h1    = ws;              ws += (long)N * F1;     // x@W1
    float* acc1  = ws;              ws += (long)N * F1;     // layer-1 agg, then elu'd in place
    float* h2    = ws;              ws += (long)N * OUT_C;  // hmid@W2
    float* acc2  = ws;              ws += (long)N * OUT_C;  // layer-2 agg
    float* asrc1 = ws;              ws += (long)N * HEADS;
    float* adst1 = ws;              ws += (long)N * HEADS;
    float* amax1 = ws;              ws += (long)N * HEADS;
    float* den1  = ws;              ws += (long)N * HEADS;
    float* asrc2 = ws;              ws += N;
    float* adst2 = ws;              ws += N;
    float* amax2 = ws;              ws += N;
    float* den2  = ws;              ws += N;

    const int BT = 256;
    auto blocks = [](long n, int bt) { return (int)((n + bt - 1) / bt); };

    // ---- init scratch (every call; ws is not re-poisoned between replays) ----
    gat_fill<<<blocks((long)N * F1, BT), BT, 0, stream>>>(acc1, 0.f, N * F1);
    gat_fill<<<blocks((long)N * OUT_C, BT), BT, 0, stream>>>(acc2, 0.f, N * OUT_C);
    gat_fill<<<blocks((long)N * HEADS, BT), BT, 0, stream>>>(den1, 0.f, N * HEADS);
    gat_fill<<<blocks(N, BT), BT, 0, stream>>>(den2, 0.f, N);
    gat_fill<<<blocks((long)N * HEADS, BT), BT, 0, stream>>>(amax1, -3.0e38f, N * HEADS);
    gat_fill<<<blocks(N, BT), BT, 0, stream>>>(amax2, -3.0e38f, N);

    // ---- layer 1 ----
    {   // h1 = x @ W1  (fp32 WMMA; 16x64 strip per wave)
        dim3 grid(N / 16, F1 / 64);
        gat_gemm_wmma_f32<4><<<grid, 32, 0, stream>>>(x, W1, h1, IN_C, F1);
    }
    gat_node_alpha<<<blocks((long)N * HEADS, BT), BT, 0, stream>>>(
        h1, att_src1, att_dst1, asrc1, adst1, N, HEADS, HID);
    gat_edge_max<<<blocks(EE, BT), BT, 0, stream>>>(ei, E, N, asrc1, adst1, amax1, HEADS);
    gat_edge_sum<<<blocks(EE, BT), BT, 0, stream>>>(ei, E, N, asrc1, adst1, amax1, den1, HEADS);
    gat_edge_scatter<<<blocks((long)EE * 32, BT), BT, 0, stream>>>(
        ei, E, N, asrc1, adst1, amax1, den1, h1, acc1, HEADS, HID);
    gat_elu_bias<<<blocks((long)N * F1, BT), BT, 0, stream>>>(acc1, b1, N, F1);

    // ---- layer 2 (H=1, C=64) ----
    {   // h2 = elu(acc1) @ W2  (16x64 strip per wave)
        dim3 grid(N / 16, OUT_C / 64);
        gat_gemm_wmma_f32<4><<<grid, 32, 0, stream>>>(acc1, W2, h2, F1, OUT_C);
    }
    gat_node_alpha<<<blocks(N, BT), BT, 0, stream>>>(
        h2, att_src2, att_dst2, asrc2, adst2, N, 1, OUT_C);
    gat_edge_max<<<blocks(EE, BT), BT, 0, stream>>>(ei, E, N, asrc2, adst2, amax2, 1);
    gat_edge_sum<<<blocks(EE, BT), BT, 0, stream>>>(ei, E, N, asrc2, adst2, amax2, den2, 1);
    gat_edge_scatter<<<blocks((long)EE * 32, BT), BT, 0, stream>>>(
        ei, E, N, asrc2, adst2, amax2, den2, h2, acc2, 1, OUT_C);
    gat_final_bias<<<blocks((long)N * OUT_C, BT), BT, 0, stream>>>(acc2, b2, out, N, OUT_C);
}